// IFNode_21483426414797
// MI455X (gfx1250) — compile-verified
//
#include <hip/hip_runtime.h>

// IF-neuron forward scan, MI455X (gfx1250, wave32).
// X: [B=64, T=32, N=16384] f32  ->  spikes S: [B, T, N] f32.
// Pure HBM-streaming kernel: 268 MB total traffic, ~11.5 us roofline @ 23.3 TB/s.

typedef __attribute__((ext_vector_type(4))) float v4f;

constexpr int kB  = 64;
constexpr int kT  = 32;
constexpr int kN  = 16384;
constexpr int kNV = kN / 4;          // float4 lanes per (b, t) row = 4096 (power of 2)
constexpr int kThreads = kB * kNV;   // 262144 total threads
constexpr int kBlock   = 256;        // 8 wave32 per block

__global__ __launch_bounds__(kBlock)
void if_scan_kernel(const float* __restrict__ X, float* __restrict__ S) {
    const int idx = blockIdx.x * kBlock + threadIdx.x;   // exact grid, no tail
    const int b   = idx >> 12;                           // idx / kNV
    const int v   = idx & (kNV - 1);                     // idx % kNV

    const size_t base = (size_t)b * (size_t)(kT * kN) + (size_t)v * 4u;
    const v4f* __restrict__ xp = reinterpret_cast<const v4f*>(X + base);
    v4f*       __restrict__ sp = reinterpret_cast<v4f*>(S + base);

    v4f mem;
    mem.x = 0.0f; mem.y = 0.0f; mem.z = 0.0f; mem.w = 0.0f;

#pragma unroll
    for (int t = 0; t < kT; ++t) {
        // timestep stride = kNV float4 = 64 KiB; 31*64KiB fits b128 imm offset,
        // so these become global_load_b128 [base + imm] th:TH_LOAD_NT.
        v4f x = __builtin_nontemporal_load(xp + (size_t)t * kNV);

        mem.x += x.x; mem.y += x.y; mem.z += x.z; mem.w += x.w;

        v4f s;
        // spike: mem > V_TH (==1.0); reset: mem -> 0 on spike. One v_cmp + two
        // v_cndmask per element, VCC shared between spike value and reset.
        s.x = (mem.x > 1.0f) ? 1.0f : 0.0f;  mem.x = (mem.x > 1.0f) ? 0.0f : mem.x;
        s.y = (mem.y > 1.0f) ? 1.0f : 0.0f;  mem.y = (mem.y > 1.0f) ? 0.0f : mem.y;
        s.z = (mem.z > 1.0f) ? 1.0f : 0.0f;  mem.z = (mem.z > 1.0f) ? 0.0f : mem.z;
        s.w = (mem.w > 1.0f) ? 1.0f : 0.0f;  mem.w = (mem.w > 1.0f) ? 0.0f : mem.w;

        __builtin_nontemporal_store(s, sp + (size_t)t * kNV);
    }
}

extern "C" void kernel_launch(void* const* d_in, const int* in_sizes, int n_in,
                              void* d_out, int out_size, void* d_ws, size_t ws_size,
                              hipStream_t stream) {
    const float* X = (const float*)d_in[0];
    float*       S = (float*)d_out;
    dim3 grid(kThreads / kBlock);   // 1024 blocks
    dim3 block(kBlock);             // 256 threads = 8 waves (wave32)
    if_scan_kernel<<<grid, block, 0, stream>>>(X, S);
}